// ArcfaceLoss_19945828122873
// MI455X (gfx1250) — compile-verified
//
#include <hip/hip_runtime.h>
#include <hip/hip_bf16.h>
#include <math.h>

// ---------------------------------------------------------------------------
// ArcFace loss, MI455X (gfx1250).
// Bandwidth-bound streaming reduction: 409.6 MB read, ~17.6us floor @23.3TB/s.
// CDNA5 paths used:
//   * global_load_async_to_lds_b128 (ASYNCcnt) triple-buffered pipeline
//   * s_wait_asynccnt split dependency counters
//   * wave32 shuffle reductions
// 4 blocks per row (2048 blocks, 32K waves) so enough bytes are in flight to
// cover HBM latency (23.3 TB/s x ~1us ~ 23 MB latency-BW product).
// No matmul structure -> WMMA not applicable.
// ---------------------------------------------------------------------------

namespace {

constexpr int   kB       = 512;
constexpr int   kC       = 100000;
constexpr int   kTPB     = 512;                  // 16 waves (wave32)
constexpr int   kNV4     = kC / 4;               // 25000 float4 per row
constexpr int   kSPLIT   = 4;                    // blocks per row
constexpr int   kV4Part  = kNV4 / kSPLIT;        // 6250 float4 per block
constexpr int   kNCHUNK  = (kV4Part + kTPB - 1) / kTPB;  // 13 chunks
constexpr float kSCALE   = 64.0f;
constexpr float kM2      = 0.5f;
constexpr float kTHRESH  = -0.87758256189f;      // cos(pi - 0.5)

__device__ __forceinline__ float arc_logit(float v) {
  // margin1==1, margin3==0 branch: theta = cos(acos(v) + m2)
  float th = cosf(acosf(v) + kM2);
  float tv = (v > kTHRESH) ? th : (-2.0f - th);  // THETA_MIN = -2
  return kSCALE * tv;
}

// online softmax update: (m, s) <- single logit l
__device__ __forceinline__ void osm_update(float l, float& m, float& s) {
  if (l > m) {
    s = s * __expf(m - l) + 1.0f;
    m = l;
  } else {
    s += __expf(l - m);
  }
}

// exact merge of two online-softmax partial states
__device__ __forceinline__ void osm_combine(float& m, float& s, float om, float os) {
  float nm = fmaxf(m, om);
  s = s * __expf(m - nm) + os * __expf(om - nm);
  m = nm;
}

} // namespace

__global__ __launch_bounds__(kTPB)
void arcface_partial_kernel(const float* __restrict__ y_true,
                            const float* __restrict__ logits,
                            float* __restrict__ pm,   // [kB*kSPLIT]
                            float* __restrict__ ps,   // [kB*kSPLIT]
                            float* __restrict__ pl) { // [kB*kSPLIT]
  __shared__ float4 sx[3][kTPB];   // staged norm_logits (8 KB x3)
  __shared__ float4 st[3][kTPB];   // staged y_true      (8 KB x3)
  __shared__ float  red_m[kTPB / 32];
  __shared__ float  red_s[kTPB / 32];
  __shared__ float  red_l[kTPB / 32];

  const int row  = blockIdx.x;
  const int part = blockIdx.y;
  const int tid  = threadIdx.x;
  const int base = part * kV4Part;            // first float4 index of our slice
  const int end  = base + kV4Part;            // one past last (exact: 25000/4)

  const float* __restrict__ xrow = logits + (size_t)row * kC;
  const float* __restrict__ trow = y_true + (size_t)row * kC;

  // Issue async global->LDS b128 loads for chunk c into buffer b.
  // Tail lanes clamp to a valid address so EVERY wave always issues exactly
  // 2 async ops per chunk -> uniform s_wait_asynccnt bookkeeping.
  auto issue = [&](int c, int b) {
    int v4 = base + c * kTPB + tid;
    if (v4 >= end) v4 = end - 1;
    unsigned long long gx = (unsigned long long)(const void*)(xrow + 4 * v4);
    unsigned long long gt = (unsigned long long)(const void*)(trow + 4 * v4);
    unsigned lx = (unsigned)(unsigned long long)(const void*)&sx[b][tid];
    unsigned lt = (unsigned)(unsigned long long)(const void*)&st[b][tid];
    asm volatile("global_load_async_to_lds_b128 %0, %1, off"
                 :: "v"(lx), "v"(gx) : "memory");
    asm volatile("global_load_async_to_lds_b128 %0, %1, off"
                 :: "v"(lt), "v"(gt) : "memory");
  };

  // Prime a depth-3 pipeline: two chunks in flight before first compute.
  issue(0, 0);
  issue(1, 1);

  float m = -INFINITY;   // running max of scaled logits over our slice
  float s = 0.0f;        // running sum exp(l - m)
  float L = -INFINITY;   // modified label logit (only the label lane sets it)

  for (int c = 0; c < kNCHUNK; ++c) {
    const int b = c % 3;
    if (c + 2 < kNCHUNK) {
      issue(c + 2, (c + 2) % 3);
      // outstanding: chunks c, c+1, c+2 = 6 ops; wait until chunk c's 2 land
      asm volatile("s_wait_asynccnt 0x4" ::: "memory");
    } else if (c + 1 < kNCHUNK) {
      asm volatile("s_wait_asynccnt 0x2" ::: "memory");
    } else {
      asm volatile("s_wait_asynccnt 0x0" ::: "memory");
    }
    __syncthreads();   // all waves' chunk-c data visible in LDS

    const int v4 = base + c * kTPB + tid;
    if (v4 < end) {
      const float4 xv = sx[b][tid];
      const float4 tv = st[b][tid];
      const float xs[4] = {xv.x, xv.y, xv.z, xv.w};
      const float ts[4] = {tv.x, tv.y, tv.z, tv.w};
#pragma unroll
      for (int k = 0; k < 4; ++k) {
        float l = kSCALE * xs[k];
        if (ts[k] != 0.0f) {       // one-hot hit: substitute margin logit
          l = arc_logit(xs[k]);
          L = l;
        }
        osm_update(l, m, s);
      }
    }
    __syncthreads();   // protect buffer b from early refill (c-1 ~ c+2 mod 3)
  }

  // ---- wave32 reduction of (m, s) and max-reduction of L ----
  const int lane = tid & 31;
  const int wave = tid >> 5;
#pragma unroll
  for (int off = 16; off; off >>= 1) {
    float om = __shfl_xor(m, off, 32);
    float os = __shfl_xor(s, off, 32);
    osm_combine(m, s, om, os);
    L = fmaxf(L, __shfl_xor(L, off, 32));
  }
  if (lane == 0) { red_m[wave] = m; red_s[wave] = s; red_l[wave] = L; }
  __syncthreads();

  if (wave == 0) {
    constexpr int NW = kTPB / 32;  // 16
    float mm = (lane < NW) ? red_m[lane] : -INFINITY;
    float ss = (lane < NW) ? red_s[lane] : 0.0f;
    float ll = (lane < NW) ? red_l[lane] : -INFINITY;
#pragma unroll
    for (int off = 16; off; off >>= 1) {
      float om = __shfl_xor(mm, off, 32);
      float os = __shfl_xor(ss, off, 32);
      osm_combine(mm, ss, om, os);
      ll = fmaxf(ll, __shfl_xor(ll, off, 32));
    }
    if (lane == 0) {
      const int idx = row * kSPLIT + part;
      pm[idx] = mm;
      ps[idx] = ss;
      pl[idx] = ll;
    }
  }
}

// Merge the kSPLIT partials per row, compute per-row loss, reduce to the mean.
__global__ __launch_bounds__(kTPB)
void arcface_combine_kernel(const float* __restrict__ pm,
                            const float* __restrict__ ps,
                            const float* __restrict__ pl,
                            float* __restrict__ out) {
  __shared__ float red[kTPB / 32];
  const int r    = threadIdx.x;     // one row per thread; kB == kTPB == 512
  const int lane = r & 31;
  const int wave = r >> 5;

  float m = pm[r * kSPLIT + 0];
  float s = ps[r * kSPLIT + 0];
  float L = pl[r * kSPLIT + 0];
#pragma unroll
  for (int p = 1; p < kSPLIT; ++p) {
    osm_combine(m, s, pm[r * kSPLIT + p], ps[r * kSPLIT + p]);
    L = fmaxf(L, pl[r * kSPLIT + p]);
  }
  // loss_row = logsumexp - label_logit = m + log(s) - L
  float v = m + __logf(s) - L;

#pragma unroll
  for (int off = 16; off; off >>= 1) v += __shfl_xor(v, off, 32);
  if (lane == 0) red[wave] = v;
  __syncthreads();
  if (wave == 0) {
    float w = (lane < kTPB / 32) ? red[lane] : 0.0f;
#pragma unroll
    for (int off = 16; off; off >>= 1) w += __shfl_xor(w, off, 32);
    if (lane == 0) out[0] = w * (1.0f / (float)kB);
  }
}

extern "C" void kernel_launch(void* const* d_in, const int* in_sizes, int n_in,
                              void* d_out, int out_size, void* d_ws, size_t ws_size,
                              hipStream_t stream) {
  const float* y_true = (const float*)d_in[0];   // [512, 100000] one-hot fp32
  const float* logits = (const float*)d_in[1];   // [512, 100000] fp32
  float* pm  = (float*)d_ws;                     // [512*4] partial max
  float* ps  = pm + kB * kSPLIT;                 // [512*4] partial sumexp
  float* pl  = ps + kB * kSPLIT;                 // [512*4] partial label logit
  float* out = (float*)d_out;                    // scalar fp32

  dim3 grid(kB, kSPLIT);
  arcface_partial_kernel<<<grid, kTPB, 0, stream>>>(y_true, logits, pm, ps, pl);
  arcface_combine_kernel<<<1, kTPB, 0, stream>>>(pm, ps, pl, out);
}